// ConvCapsLayer3D_70566312673301
// MI455X (gfx1250) — compile-verified
//
#include <hip/hip_runtime.h>
#include <hip/hip_bf16.h>

// ConvCapsLayer3D for MI455X (gfx1250).
// conv -> WMMA f32 16x16x4 GEMM over output channels; routing -> bandwidth
// kernels with TDM async LDS staging. All f32 (problem is HBM-bound, not
// compute-bound: ~1.6GB traffic ~ 70-90us at 23.3 TB/s).

typedef __attribute__((ext_vector_type(2)))  float v2f;
typedef __attribute__((ext_vector_type(8)))  float v8f;
typedef __attribute__((ext_vector_type(4)))  unsigned int u32x4;
typedef __attribute__((ext_vector_type(8)))  int i32x8;
typedef __attribute__((ext_vector_type(4)))  int i32x4;

#define B_     16
#define CH_I   32
#define N_I    4
#define CH_J   32
#define N_J    4
#define HW_    32
#define OC_    128            // CH_J*N_J
#define TAPS   27

#define XC_ELEMS   (67108864ull)   // B*H*W*CH_J*N_J*CH_I
#define BL_ELEMS   (16777216ull)   // B*H*W*CH_J*CH_I

#if defined(__has_builtin)
#if __has_builtin(__builtin_amdgcn_tensor_load_to_lds)
#define USE_TDM 1
#endif
#endif
#ifndef USE_TDM
#define USE_TDM 0
#endif

// ---------------------------------------------------------------------------
// Conv kernel: per block = (b, d, 4 h-rows). 8 waves: wave = (h_local, w half).
// Each wave: M = 16 consecutive w positions, N = 128 ocs (8 tiles of 16),
// K = 27 taps padded to 28 -> 7 x V_WMMA_F32_16X16X4_F32 per oc tile.
// ---------------------------------------------------------------------------
__global__ __launch_bounds__(256) void conv_wmma_caps(
    const float* __restrict__ x,      // [B][128][32][32] (ch_i*4+n_i planes)
    const float* __restrict__ wgt,    // [128][27]
    const float* __restrict__ bias,   // [128]
    float* __restrict__ xc)           // [b][h][w][ch_j][n_j][ch_i]
{
    const int d  = blockIdx.x;        // ch_i of output (depth position)
    const int hb = blockIdx.y;        // group of 4 h rows
    const int b  = blockIdx.z;
    const int tid  = threadIdx.x;
    const int lane = tid & 31;
    const int wv   = tid >> 5;

    __shared__ float wlds[28 * 128];          // [tap][oc], tap 27 zero-padded
    __shared__ float slds[3 * 6 * 36];        // [kd][h0-1..h0+4][w -1..32] padded

    for (int i = tid; i < 28 * 128; i += 256) {
        int t = i >> 7, oc = i & 127;
        wlds[i] = (t < TAPS) ? wgt[oc * TAPS + t] : 0.0f;
    }
    const int h0 = hb * 4;
    for (int i = tid; i < 3 * 6 * 34; i += 256) {
        int kd = i / (6 * 34); int r = i % (6 * 34);
        int j = r / 34; int xw = r % 34;
        int hh = h0 - 1 + j; int ww = xw - 1;
        int c  = 4 * d + kd;                           // input depth plane
        float v = 0.0f;
        if (hh >= 0 && hh < HW_ && ww >= 0 && ww < HW_)
            v = x[((size_t)b * 128 + c) * 1024 + hh * 32 + ww];
        slds[(kd * 6 + j) * 36 + xw] = v;
    }
    __syncthreads();

    const int hl    = wv >> 1;          // 0..3 local h
    const int w0    = (wv & 1) * 16;    // w tile base
    const int m     = lane & 15;        // M index held by this lane
    const int khalf = lane >> 4;        // K half (A/B layout split)

    v8f acc[8];
#pragma unroll
    for (int n = 0; n < 8; ++n) acc[n] = v8f{0, 0, 0, 0, 0, 0, 0, 0};

#pragma unroll
    for (int c = 0; c < 7; ++c) {                       // K chunks of 4
        v2f a;
#pragma unroll
        for (int r = 0; r < 2; ++r) {
            int t = 4 * c + khalf * 2 + r;
            float av = 0.0f;
            if (t < TAPS) {
                int kd = t / 9, kh = (t % 9) / 3, kw = t % 3;
                av = slds[(kd * 6 + (hl + kh)) * 36 + (w0 + m + kw)];
            }
            a[r] = av;
        }
#pragma unroll
        for (int n = 0; n < 8; ++n) {                   // oc tiles
            v2f bb;
#pragma unroll
            for (int r = 0; r < 2; ++r) {
                int t = 4 * c + khalf * 2 + r;
                bb[r] = wlds[t * 128 + n * 16 + m];
            }
            acc[n] = __builtin_amdgcn_wmma_f32_16x16x4_f32(
                false, a, false, bb, (short)0, acc[n], false, false);
        }
    }

    // C layout: lane l, vgpr r -> M = r + 8*(l>>4), N = l&15
    const int h = h0 + hl;
#pragma unroll
    for (int n = 0; n < 8; ++n) {
        int oc  = n * 16 + m;
        float bv = bias[oc];
        int chj = oc >> 2, nj = oc & 3;
#pragma unroll
        for (int r = 0; r < 8; ++r) {
            int ww = w0 + r + khalf * 8;
            size_t idx = ((((size_t)((b * 32 + h) * 32 + ww)) * 32 + chj) * 4 + nj) * 32 + d;
            xc[idx] = acc[n][r] + bv;
        }
    }
}

// ---------------------------------------------------------------------------
// Softmax statistics per (b, ch_i): max and sum(exp) over 32768 logits.
// ---------------------------------------------------------------------------
__global__ __launch_bounds__(1024) void route_reduce(
    const float* __restrict__ Bl, float* __restrict__ mOut, float* __restrict__ dOut)
{
    const int b  = blockIdx.x;
    const int ci = threadIdx.x & 31;
    const int g  = threadIdx.x >> 5;          // 0..31
    const float* base = Bl + (size_t)b * 32768 * 32;

    __shared__ float red[32][33];

    float mx = -3.0e38f;
    for (int p = g; p < 32768; p += 32)
        mx = fmaxf(mx, base[(size_t)p * 32 + ci]);
    red[g][ci] = mx;
    __syncthreads();
    for (int s = 16; s > 0; s >>= 1) {
        if (g < s) red[g][ci] = fmaxf(red[g][ci], red[g + s][ci]);
        __syncthreads();
    }
    const float mv = red[0][ci];
    __syncthreads();

    float sum = 0.0f;
    for (int p = g; p < 32768; p += 32)
        sum += __expf(base[(size_t)p * 32 + ci] - mv);
    red[g][ci] = sum;
    __syncthreads();
    for (int s = 16; s > 0; s >>= 1) {
        if (g < s) red[g][ci] += red[g + s][ci];
        __syncthreads();
    }
    if (g == 0) {
        mOut[b * 32 + ci] = mv;
        dOut[b * 32 + ci] = red[0][ci];
    }
}

// ---------------------------------------------------------------------------
// Routing update: thread = one (b,h,w,ch_j). xc tile (256*128 f32 = 128KB)
// staged into LDS by the Tensor Data Mover (one descriptor per block).
// ITER 0: uniform softmax (Bl==0), write Bl. ITER 1: read/write Bl.
// ITER 2: write final output [B][ch_j][n_j][H][W].
// ---------------------------------------------------------------------------
template <int ITER>
__global__ __launch_bounds__(256) void route_update(
    const float* __restrict__ xc, float* __restrict__ Bl,
    const float* __restrict__ mIn, const float* __restrict__ dIn,
    float* __restrict__ out)
{
    const int t = threadIdx.x;
    const size_t gt = (size_t)blockIdx.x * 256 + t;   // ((b*1024 + hw)*32 + chj)
    const int b = (int)(gt >> 15);

#if USE_TDM
    __shared__ float tile[256 * 128];                 // 128 KB (<=320KB/WG on CDNA5)
    if (t < 32) {
        const float* gsrc = xc + (size_t)blockIdx.x * 256 * 128;
        unsigned long long ga = (unsigned long long)(uintptr_t)gsrc;
        unsigned lds = (unsigned)(uintptr_t)&tile[0];
        const unsigned elems = 256u * 128u;           // 32768 dwords
        u32x4 g0;
        g0[0] = 1u;                                   // count=1, user descriptor
        g0[1] = lds;                                  // lds_addr (bytes)
        g0[2] = (unsigned)(ga & 0xFFFFFFFFu);         // global_addr[31:0]
        g0[3] = ((unsigned)(ga >> 32) & 0x01FFFFFFu) | (2u << 30); // addr[56:32] | type=2
        i32x8 g1;
        g1[0] = (int)(2u << 16);                      // data_size = 4B
        g1[1] = (int)((elems & 0xFFFFu) << 16);       // tensor_dim0[15:0]
        g1[2] = (int)(((elems >> 16) & 0xFFFFu) | (1u << 16)); // td0[31:16] | td1=1
        g1[3] = (int)((elems & 0xFFFFu) << 16);       // tile_dim0 = 32768
        g1[4] = 1;                                    // tile_dim1 = 1
        g1[5] = (int)elems;                           // tensor_dim0_stride
        g1[6] = 0; g1[7] = 0;
        i32x4 g2; g2[0] = 0; g2[1] = 0; g2[2] = 0; g2[3] = 0;
        i32x4 g3; g3[0] = 0; g3[1] = 0; g3[2] = 0; g3[3] = 0;
        i32x8 g4; g4[0] = 0; g4[1] = 0; g4[2] = 0; g4[3] = 0;
                  g4[4] = 0; g4[5] = 0; g4[6] = 0; g4[7] = 0;
        __builtin_amdgcn_tensor_load_to_lds(g0, g1, g2, g3, g4, 0);
        __builtin_amdgcn_s_wait_tensorcnt(0);
    }
    __syncthreads();
    const float* xl = &tile[t * 128];                 // [n_j][ch_i] for this (hw,chj)
#else
    const float* xl = xc + gt * 128;
#endif

    // Pass 1: S[nj] = sum_i k_i * xc[nj][i]
    float S[4] = {0.0f, 0.0f, 0.0f, 0.0f};
    for (int i = 0; i < 32; ++i) {
        float ki;
        if (ITER == 0) {
            ki = 1.0f / 32768.0f;
        } else {
            float bl = Bl[gt * 32 + i];
            ki = __expf(bl - mIn[b * 32 + i]) / dIn[b * 32 + i];
        }
#pragma unroll
        for (int nj = 0; nj < 4; ++nj) S[nj] += ki * xl[nj * 32 + i];
    }

    // squash: s * |s| / (1 + s^2 + eps)
    float Shat[4];
#pragma unroll
    for (int nj = 0; nj < 4; ++nj) {
        float s = S[nj];
        float nrm = fabsf(s);
        Shat[nj] = nrm / (1.0f + nrm * nrm + 1e-20f) * s;
    }

    if (ITER < 2) {
        for (int i = 0; i < 32; ++i) {
            float acc = (ITER == 0) ? 0.0f : Bl[gt * 32 + i];
#pragma unroll
            for (int nj = 0; nj < 4; ++nj) acc += Shat[nj] * xl[nj * 32 + i];
            Bl[gt * 32 + i] = acc;
        }
    } else {
        const int chj = (int)(gt & 31);
        const int hw  = (int)((gt >> 5) & 1023);
        const int h = hw >> 5, w = hw & 31;
#pragma unroll
        for (int nj = 0; nj < 4; ++nj)
            out[((((size_t)b * 32 + chj) * 4 + nj) * 32 + h) * 32 + w] = Shat[nj];
    }
}

// ---------------------------------------------------------------------------
extern "C" void kernel_launch(void* const* d_in, const int* in_sizes, int n_in,
                              void* d_out, int out_size, void* d_ws, size_t ws_size,
                              hipStream_t stream)
{
    const float* x    = (const float*)d_in[0];   // (16,32,4,32,32)
    const float* wgt  = (const float*)d_in[1];   // (128,1,3,3,3)
    const float* bias = (const float*)d_in[2];   // (128,)
    float* out = (float*)d_out;

    float* ws  = (float*)d_ws;
    float* xc  = ws;                             // 256 MB
    float* Bl  = ws + XC_ELEMS;                  // 64 MB
    float* mW  = Bl + BL_ELEMS;                  // 512 f32
    float* dW  = mW + 512;                       // 512 f32

    dim3 gc(32, 8, 16);                          // (d, h/4, b)
    conv_wmma_caps<<<gc, dim3(256), 0, stream>>>(x, wgt, bias, xc);

    const int upd_blocks = (int)((16ull * 32 * 32 * 32) / 256); // 2048
    route_update<0><<<upd_blocks, 256, 0, stream>>>(xc, Bl, mW, dW, out);
    route_reduce<<<16, 1024, 0, stream>>>(Bl, mW, dW);
    route_update<1><<<upd_blocks, 256, 0, stream>>>(xc, Bl, mW, dW, out);
    route_reduce<<<16, 1024, 0, stream>>>(Bl, mW, dW);
    route_update<2><<<upd_blocks, 256, 0, stream>>>(xc, Bl, mW, dW, out);
}